// ReservoirCell_27453430956032
// MI455X (gfx1250) — compile-verified
//
#include <hip/hip_runtime.h>
#include <math.h>

// ---------------------------------------------------------------------------
// Echo-state network forward for MI455X (gfx1250, wave32).
// Recurrent GEMM on V_WMMA_F32_16X16X4_F32 (fp32 in / fp32 acc, matching
// reference numerics). Persistent kernel; sequential timesteps synchronized
// per independent 16-batch-row M-group via an atomic barrier. h_prev A-tile
// staged into LDS with GLOBAL_LOAD_ASYNC_TO_LDS_B128 (ASYNCcnt), overlapped
// with the fused input-projection WMMA loop.
// ---------------------------------------------------------------------------

typedef __attribute__((ext_vector_type(2))) float v2f;
typedef __attribute__((ext_vector_type(8))) float v8f;

#define B_   64
#define T_   512
#define D_   128
#define U_   2048
#define LEAK 0.9f

#define LDS_STRIDE 2052   // 2048 + 4 dword pad: 2052 % 64 == 4 -> conflict-free

// workspace layout (bytes); total ~19 MB
#define BAR_OFF  0                          // 16 uints: arrive[4], gen[4]
#define WIN_OFF  256                        // dense W_in: 128*2048 f32 (1 MB)
#define WINP_OFF (256 + D_*U_*4)            // packed W_in fragments (1 MB)
#define WP_OFF   (256 + 2*D_*U_*4)          // packed W fragments (16 MB)

// ---- build dense W_in deterministically (one thread per row, serial adds,
// ---- same duplicate-sum order as the reference scatter-add) ----
__global__ void esn_scatter_win(const float* __restrict__ vals,
                                const int* __restrict__ cols,
                                float* __restrict__ win) {
  int r = threadIdx.x;
  if (r < D_) {
    for (int c = 0; c < 10; ++c) {
      win[r * U_ + cols[r * 10 + c]] += vals[r * 10 + c];
    }
  }
}

// ---- pack W and W_in into per-lane WMMA B-fragment order ----
// Fragment p = (((ni*KSTEPS)+ks)*32 + lane)*2 + j holds
//   W[4*ks + 2*(lane>=16) + j][ni*16 + (lane&15)]
// so the hot loop does one coalesced b64 load per lane per k-step.
__global__ void esn_pack(const float* __restrict__ W,
                         const float* __restrict__ win,
                         float* __restrict__ wp,
                         float* __restrict__ winp) {
  int p = blockIdx.x * blockDim.x + threadIdx.x;
  const int NPW = U_ * U_;
  const int NPI = D_ * U_;
  if (p < NPW) {
    int j = p & 1, lane = (p >> 1) & 31, ks = (p >> 6) & 511, ni = p >> 15;
    int row = 4 * ks + ((lane >> 4) << 1) + j;
    int col = ni * 16 + (lane & 15);
    wp[p] = W[(size_t)row * U_ + col];
  } else if (p < NPW + NPI) {
    int q = p - NPW;
    int j = q & 1, lane = (q >> 1) & 31, ks = (q >> 6) & 31, ni = q >> 11;
    int row = 4 * ks + ((lane >> 4) << 1) + j;
    int col = ni * 16 + (lane & 15);
    winp[q] = win[(size_t)row * U_ + col];
  }
}

// async memory -> LDS copy, 16B per lane, tracked by ASYNCcnt
__device__ __forceinline__ void async_to_lds_b128(unsigned lds_off,
                                                  const void* gptr) {
  asm volatile("global_load_async_to_lds_b128 %0, %1, off"
               :: "v"(lds_off), "v"(gptr)
               : "memory");
}

__device__ __forceinline__ void wait_asynccnt0() {
  asm volatile("s_wait_asynccnt 0" ::: "memory");
}

// ---- persistent reservoir scan ----
// Grid: 32 blocks x 256 threads (8 waves). Block bid: group g = bid>>3
// (batch rows g*16..g*16+15), wg = bid&7. Each wave owns two 16-col N-tiles.
// Per step, per k-step: 1 ds_load (A from LDS) + 2 global b64 (packed W)
// + 2 fp32 WMMAs.
__launch_bounds__(256)
__global__ void esn_main(const float* __restrict__ x,
                         const float* __restrict__ bias,
                         const float* __restrict__ wp,
                         const float* __restrict__ winp,
                         float* __restrict__ out,
                         unsigned int* __restrict__ bar) {
  __shared__ float ldsA[16 * LDS_STRIDE];   // 131,328 B (<320 KB/WGP)

  const int tid  = threadIdx.x;
  const int wave = tid >> 5;
  const int lane = tid & 31;
  const int bid  = blockIdx.x;
  const int g    = bid >> 3;            // independent M-group (0..3)
  const int wg   = bid & 7;
  const int m0   = g * 16;
  const int ni0  = wg * 16 + wave * 2;  // this wave's two N-tiles
  const int lc   = lane & 15;
  const int kh   = (lane >> 4) << 1;    // A/B fragment K offset: 0 or 2
  const int mh   = (lane >> 4) << 3;    // C fragment M offset: 0 or 8

  const int n0 = ni0 * 16 + lc;
  const int n1 = n0 + 16;
  const float bn0 = bias[n0];
  const float bn1 = bias[n1];

  const float* wp0 = wp + (size_t)ni0 * 512 * 64 + lane * 2;
  const float* wp1 = wp0 + 512 * 64;
  const float* wi0 = winp + (size_t)ni0 * 32 * 64 + lane * 2;
  const float* wi1 = wi0 + 32 * 64;
  const float* lA  = ldsA + lc * LDS_STRIDE + kh;   // A fragment base in LDS

  unsigned int* arrive = bar + g;
  unsigned int* genp   = bar + 8 + g;

  for (int t = 0; t < T_; ++t) {
    // ---- stage h_prev[16 x 2048] into LDS asynchronously (zeros at t==0);
    // ---- overlapped with the input-projection WMMA loop below ----
    if (t > 0) {
#pragma unroll 4
      for (int c = tid; c < 16 * 512; c += 256) {          // 16 rows x 512 float4
        int row = c >> 9;
        int cc  = (c & 511) << 2;
        unsigned loff = (unsigned)(size_t)&ldsA[row * LDS_STRIDE + cc];
        const float* gsrc = out + ((size_t)(m0 + row) * T_ + (t - 1)) * U_ + cc;
        async_to_lds_b128(loff, gsrc);
      }
    } else {
      float4 z = make_float4(0.f, 0.f, 0.f, 0.f);
      for (int c = tid; c < 16 * 512; c += 256) {
        int row = c >> 9;
        int cc  = (c & 511) << 2;
        *(float4*)(&ldsA[row * LDS_STRIDE + cc]) = z;
      }
    }

    v8f acc0 = {};
    v8f acc1 = {};

    // fused input projection: u_t = x[b,t,:] @ W_in  (K = 128, no LDS use)
    const float* xA = x + ((size_t)(m0 + lc) * T_ + t) * D_ + kh;
#pragma unroll 8
    for (int ks = 0; ks < D_ / 4; ++ks) {
      v2f a  = *(const v2f*)(xA + 4 * ks);
      v2f b0 = *(const v2f*)(wi0 + ks * 64);
      v2f b1 = *(const v2f*)(wi1 + ks * 64);
      acc0 = __builtin_amdgcn_wmma_f32_16x16x4_f32(false, a, false, b0,
                                                   (short)0, acc0, false, false);
      acc1 = __builtin_amdgcn_wmma_f32_16x16x4_f32(false, a, false, b1,
                                                   (short)0, acc1, false, false);
    }

    // own async fills done; then make all waves' fills visible
    wait_asynccnt0();
    __syncthreads();

    // recurrent term: h_prev @ W  (K = 2048); A fragments from LDS
    if (t > 0) {
#pragma unroll 8
      for (int ks = 0; ks < U_ / 4; ++ks) {
        v2f a  = *(const v2f*)(lA + 4 * ks);
        v2f b0 = *(const v2f*)(wp0 + (size_t)ks * 64);
        v2f b1 = *(const v2f*)(wp1 + (size_t)ks * 64);
        acc0 = __builtin_amdgcn_wmma_f32_16x16x4_f32(false, a, false, b0,
                                                     (short)0, acc0, false, false);
        acc1 = __builtin_amdgcn_wmma_f32_16x16x4_f32(false, a, false, b1,
                                                     (short)0, acc1, false, false);
      }
    }

    // epilogue: h_new = (1-LEAK)*h_prev + LEAK*tanh(acc + bias)
    // h_prev read back from the LDS stage (zeros at t == 0).
#pragma unroll
    for (int j = 0; j < 8; ++j) {
      int ml = mh + j;                           // C layout: M = j + 8*(lane>=16)
      size_t ro = ((size_t)(m0 + ml) * T_ + t) * U_;
      float hp0 = ldsA[ml * LDS_STRIDE + n0];
      float hp1 = ldsA[ml * LDS_STRIDE + n1];
      float v0 = tanhf(acc0[j] + bn0);
      float v1 = tanhf(acc1[j] + bn1);
      out[ro + n0] = hp0 * (1.0f - LEAK) + v0 * LEAK;
      out[ro + n1] = hp1 * (1.0f - LEAK) + v1 * LEAK;
    }

    // release stores -> per-M-group barrier (8 blocks) -> acquire
    __threadfence();
    __syncthreads();
    if (tid == 0) {
      unsigned cur = __hip_atomic_load(genp, __ATOMIC_RELAXED,
                                       __HIP_MEMORY_SCOPE_AGENT);
      unsigned old = __hip_atomic_fetch_add(arrive, 1u, __ATOMIC_ACQ_REL,
                                            __HIP_MEMORY_SCOPE_AGENT);
      if (old == 7u) {
        __hip_atomic_store(arrive, 0u, __ATOMIC_RELAXED,
                           __HIP_MEMORY_SCOPE_AGENT);
        __hip_atomic_fetch_add(genp, 1u, __ATOMIC_RELEASE,
                               __HIP_MEMORY_SCOPE_AGENT);
      } else {
        while (__hip_atomic_load(genp, __ATOMIC_ACQUIRE,
                                 __HIP_MEMORY_SCOPE_AGENT) == cur) {
          __builtin_amdgcn_s_sleep(2);
        }
      }
    }
    __syncthreads();
    __threadfence();
  }
}

extern "C" void kernel_launch(void* const* d_in, const int* in_sizes, int n_in,
                              void* d_out, int out_size, void* d_ws, size_t ws_size,
                              hipStream_t stream) {
  const float* x    = (const float*)d_in[0];  // [64,512,128]
  const float* kv   = (const float*)d_in[1];  // [128,10]
  const float* W    = (const float*)d_in[2];  // [2048,2048]
  const float* bias = (const float*)d_in[3];  // [2048]
  const int*   kc   = (const int*)d_in[4];    // [128,10]
  float* out = (float*)d_out;                 // [64,512,2048]

  char* ws = (char*)d_ws;                     // needs ~19 MB
  unsigned int* bar = (unsigned int*)(ws + BAR_OFF);
  float* win  = (float*)(ws + WIN_OFF);
  float* winp = (float*)(ws + WINP_OFF);
  float* wp   = (float*)(ws + WP_OFF);

  // zero barrier state + dense W_in (graph-capturable memset node)
  hipMemsetAsync(ws, 0, WINP_OFF, stream);
  esn_scatter_win<<<1, 128, 0, stream>>>(kv, kc, win);
  const int totalPack = U_ * U_ + D_ * U_;
  esn_pack<<<(totalPack + 255) / 256, 256, 0, stream>>>(W, win, wp, winp);
  esn_main<<<32, 256, 0, stream>>>(x, bias, wp, winp, out, bar);
}